// DiGCNNet_51539608034
// MI455X (gfx1250) — compile-verified
//
#include <hip/hip_runtime.h>
#include <hip/hip_bf16.h>

// Fused DiGCN forward for MI455X (gfx1250), wave32, WMMA f32 16x16x4.
//
// B=512 T=64 N=30 F_IN=512 D=256 C=7.
// One 256-thread block (8 wave32) per batch element.

typedef __attribute__((ext_vector_type(2))) float v2f;
typedef __attribute__((ext_vector_type(8))) float v8f;

#define B_SZ   512
#define T_SZ   64
#define N_SZ   30
#define FIN_SZ 512
#define D_SZ   256
#define C_SZ   7

// LDS row strides (floats), chosen for bank-conflict-free WMMA fragment access.
#define A_STRIDE   516   // real tile: 32 x 516  (66.0 KB); 516%64=4 -> A-frag b64 loads conflict-free
#define XW_STRIDE  272   // xw tile:   32 x 272  (34.8 KB); 2*272%64=32 -> B-frag reads conflict-free
#define ADJ_STRIDE 36    // adjT tile: 32 x 36   ( 4.6 KB); 36%64=36 -> A-frag b64 loads conflict-free

static __device__ __forceinline__ v8f wmma_f32_16x16x4(v2f a, v2f b, v8f c) {
    // 8-arg pattern: (neg_a, A, neg_b, B, c_mod, C, reuse_a, reuse_b)
    return __builtin_amdgcn_wmma_f32_16x16x4_f32(false, a, false, b, (short)0, c,
                                                 false, false);
}

__launch_bounds__(256)
__global__ void digcn_fused_kernel(const float* __restrict__ real,
                                   const float* __restrict__ gs,
                                   const float* __restrict__ W,
                                   const float* __restrict__ conv_bias,
                                   const float* __restrict__ pool_w,
                                   const float* __restrict__ pool_b,
                                   const float* __restrict__ head_w,
                                   const float* __restrict__ head_b,
                                   float* __restrict__ out)
{
    __shared__ float ldsA[32 * A_STRIDE];       // real_b, rows 30/31 zero
    __shared__ float ldsXW[32 * XW_STRIDE];     // xw = real_b @ W
    __shared__ float ldsAdjT[32 * ADJ_STRIDE];  // adjT[j][i] = mean_t gs[b,t,i,j], padded 0
    __shared__ float ldsScore[32];              // per-node pooled score (pre pool_b)
    __shared__ float ldsLogit[8];

    const int b    = blockIdx.x;
    const int tid  = threadIdx.x;
    const int wave = tid >> 5;
    const int lane = tid & 31;
    const int half = lane >> 4;   // 0: lanes 0-15, 1: lanes 16-31
    const int idx  = lane & 15;

    // ---- init LDS (adjT padding must be zero; score accumulators zero) ----
    for (int u = tid; u < 32 * ADJ_STRIDE; u += 256) ldsAdjT[u] = 0.0f;
    if (tid < 32) ldsScore[tid] = 0.0f;

    // ---- Phase 1: adj = mean_t graph_sigs[b]  (dominant HBM traffic) ----
    // t-outer loop: each t-slice of 900 floats read fully coalesced.
    float acc0 = 0.f, acc1 = 0.f, acc2 = 0.f, acc3 = 0.f;
    {
        const float* g = gs + (size_t)b * (T_SZ * N_SZ * N_SZ);
        for (int t = 0; t < T_SZ; ++t) {
            const float* gt = g + t * (N_SZ * N_SZ);
            acc0 += gt[tid];
            acc1 += gt[tid + 256];
            acc2 += gt[tid + 512];
            if (tid < 900 - 768) acc3 += gt[tid + 768];
        }
    }

    // ---- Phase 2a: stage real_b (30x512) into LDS, zero-pad rows 30/31 ----
    {
        const float* rb = real + (size_t)b * (N_SZ * FIN_SZ);
        for (int u = tid; u < 32 * FIN_SZ; u += 256) {
            int row = u >> 9;       // / 512
            int col = u & 511;
            float v = 0.0f;
            if (row < N_SZ) v = rb[row * FIN_SZ + col];
            ldsA[row * A_STRIDE + col] = v;
        }
    }
    __syncthreads();   // adjT zeros + ldsA visible

    // scatter adj means (transposed) into LDS
    {
        const float inv_t = 1.0f / (float)T_SZ;
        int e = tid;         int i = e / 30, j = e - i * 30;
        ldsAdjT[j * ADJ_STRIDE + i] = acc0 * inv_t;
        e = tid + 256;       i = e / 30;  j = e - i * 30;
        ldsAdjT[j * ADJ_STRIDE + i] = acc1 * inv_t;
        e = tid + 512;       i = e / 30;  j = e - i * 30;
        ldsAdjT[j * ADJ_STRIDE + i] = acc2 * inv_t;
        if (tid < 900 - 768) {
            e = tid + 768;   i = e / 30;  j = e - i * 30;
            ldsAdjT[j * ADJ_STRIDE + i] = acc3 * inv_t;
        }
    }

    // ---- Phase 2b: xw = real_b @ W via v_wmma_f32_16x16x4_f32 ----
    // Wave w handles N-tiles {w, w+8}; both M-tiles per N-tile share B-fragments.
    for (int np = 0; np < 2; ++np) {
        const int nt   = wave + np * 8;
        const int ncol = nt * 16 + idx;
        v8f acc_lo = {};   // rows 0..15
        v8f acc_hi = {};   // rows 16..31
        const float* a0 = &ldsA[idx * A_STRIDE + 2 * half];
        const float* a1 = &ldsA[(16 + idx) * A_STRIDE + 2 * half];
        for (int k0 = 0; k0 < FIN_SZ; k0 += 4) {
            v2f a_lo = *(const v2f*)(a0 + k0);          // A[idx][k0+2h .. +1]
            v2f a_hi = *(const v2f*)(a1 + k0);
            const int kr = k0 + 2 * half;
            v2f bb = { W[kr * D_SZ + ncol], W[(kr + 1) * D_SZ + ncol] };
            acc_lo = wmma_f32_16x16x4(a_lo, bb, acc_lo);
            acc_hi = wmma_f32_16x16x4(a_hi, bb, acc_hi);
        }
#pragma unroll
        for (int r = 0; r < 8; ++r) {
            ldsXW[(r + 8 * half) * XW_STRIDE + ncol]      = acc_lo[r];
            ldsXW[(16 + r + 8 * half) * XW_STRIDE + ncol] = acc_hi[r];
        }
    }
    __syncthreads();   // ldsXW + ldsAdjT ready

    // ---- Phase 3: agg = adjT @ xw; fused bias+relu+pool epilogue ----
    for (int np = 0; np < 2; ++np) {
        const int nt   = wave + np * 8;
        const int dcol = nt * 16 + idx;
        v8f acc_lo = {};   // out rows j = 0..15
        v8f acc_hi = {};   // out rows j = 16..31 (30/31 are padding)
        const float* a0 = &ldsAdjT[idx * ADJ_STRIDE + 2 * half];
        const float* a1 = &ldsAdjT[(16 + idx) * ADJ_STRIDE + 2 * half];
#pragma unroll
        for (int k0 = 0; k0 < 32; k0 += 4) {
            v2f a_lo = *(const v2f*)(a0 + k0);
            v2f a_hi = *(const v2f*)(a1 + k0);
            const int kr = k0 + 2 * half;
            v2f bb = { ldsXW[kr * XW_STRIDE + dcol],
                       ldsXW[(kr + 1) * XW_STRIDE + dcol] };
            acc_lo = wmma_f32_16x16x4(a_lo, bb, acc_lo);
            acc_hi = wmma_f32_16x16x4(a_hi, bb, acc_hi);
        }
        const float cb = conv_bias[dcol];
        const float pw = pool_w[dcol];
#pragma unroll
        for (int r = 0; r < 8; ++r) {
            float vlo = acc_lo[r] + cb;
            vlo = vlo > 0.0f ? vlo : 0.0f;
            vlo *= pw;
            vlo += __shfl_xor(vlo, 1, 32);
            vlo += __shfl_xor(vlo, 2, 32);
            vlo += __shfl_xor(vlo, 4, 32);
            vlo += __shfl_xor(vlo, 8, 32);
            if (idx == 0) atomicAdd(&ldsScore[r + 8 * half], vlo);

            float vhi = acc_hi[r] + cb;
            vhi = vhi > 0.0f ? vhi : 0.0f;
            vhi *= pw;
            vhi += __shfl_xor(vhi, 1, 32);
            vhi += __shfl_xor(vhi, 2, 32);
            vhi += __shfl_xor(vhi, 4, 32);
            vhi += __shfl_xor(vhi, 8, 32);
            if (idx == 0) atomicAdd(&ldsScore[16 + r + 8 * half], vhi);
        }
    }
    __syncthreads();

    // ---- Phase 4: head (Conv1d k=1 == dense [C,N]) + softmax ----
    if (tid < C_SZ) {
        const float pb = pool_b[0];
        float l = head_b[tid];
        for (int j = 0; j < N_SZ; ++j)
            l += head_w[tid * N_SZ + j] * (ldsScore[j] + pb);
        ldsLogit[tid] = l;
    }
    __syncthreads();
    if (tid == 0) {
        float m = ldsLogit[0];
        for (int c = 1; c < C_SZ; ++c) m = fmaxf(m, ldsLogit[c]);
        float e[C_SZ];
        float s = 0.0f;
        for (int c = 0; c < C_SZ; ++c) { e[c] = expf(ldsLogit[c] - m); s += e[c]; }
        const float inv = 1.0f / s;
        for (int c = 0; c < C_SZ; ++c) out[b * C_SZ + c] = e[c] * inv;
    }
}

extern "C" void kernel_launch(void* const* d_in, const int* in_sizes, int n_in,
                              void* d_out, int out_size, void* d_ws, size_t ws_size,
                              hipStream_t stream) {
    // setup_inputs order:
    // 0:real 1:imag(unused) 2:graph_sigs 3:W 4:conv_bias 5:pool_w 6:pool_b 7:head_w 8:head_b
    const float* real = (const float*)d_in[0];
    const float* gs   = (const float*)d_in[2];
    const float* W    = (const float*)d_in[3];
    const float* cb   = (const float*)d_in[4];
    const float* pw   = (const float*)d_in[5];
    const float* pb   = (const float*)d_in[6];
    const float* hw   = (const float*)d_in[7];
    const float* hb   = (const float*)d_in[8];
    float* out = (float*)d_out;

    digcn_fused_kernel<<<dim3(B_SZ), dim3(256), 0, stream>>>(
        real, gs, W, cb, pw, pb, hw, hb, out);
}